// myGIN_80736795230253
// MI455X (gfx1250) — compile-verified
//
#include <hip/hip_runtime.h>

// GIN 2-layer forward for MI455X (gfx1250, wave32).
// Pipeline: memset agg -> scatter(x) -> fused MLP1 (WMMA f32 16x16x4) -> h
//           memset agg -> scatter(h) -> fused MLP2 + log_softmax -> out
// Scatter is the memory-bound part (~640MB/layer of gather+atomic traffic);
// GEMMs use full-precision V_WMMA_F32_16X16X4_F32 since compute is trivial.

#define N_NODES 100000
#define N_EDGES 1250000
#define D 64
#define D_OUT 40
#define D_OUT_PAD 48
#define WAVES 8            // 8 wave32 per block = 256 threads
#define N_TILES (N_NODES / 16)   // 6250 exactly

typedef __attribute__((ext_vector_type(2))) float v2f;
typedef __attribute__((ext_vector_type(8))) float v8f;

__device__ __forceinline__ v8f wmma_f32(v2f a, v2f b, v8f c) {
  // D = A(16x4 f32) * B(4x16 f32) + C(16x16 f32)
  return __builtin_amdgcn_wmma_f32_16x16x4_f32(
      /*neg_a=*/false, a, /*neg_b=*/false, b,
      /*c_mod=*/(short)0, c, /*reuse_a=*/false, /*reuse_b=*/false);
}

// ---------------------------------------------------------------------------
// Edge scatter: one wave per edge, lane handles 2 features.
// Coalesced 256B gather from feat[src], coalesced f32 atomics into agg[dst].
// ---------------------------------------------------------------------------
__global__ __launch_bounds__(256) void gin_scatter(
    const float* __restrict__ feat, const int* __restrict__ src,
    const int* __restrict__ dst, float* __restrict__ agg) {
  int g = blockIdx.x * 256 + threadIdx.x;
  int e = g >> 5;
  if (e >= N_EDGES) return;
  int lane = g & 31;
  int s = src[e], d = dst[e];
  v2f v = *(const v2f*)(feat + (long)s * D + 2 * lane);
  float* p = agg + (long)d * D + 2 * lane;
  __hip_atomic_fetch_add(p,     v.x, __ATOMIC_RELAXED, __HIP_MEMORY_SCOPE_AGENT);
  __hip_atomic_fetch_add(p + 1, v.y, __ATOMIC_RELAXED, __HIP_MEMORY_SCOPE_AGENT);
}

// ---------------------------------------------------------------------------
// Layer 1 fused MLP: h = relu( relu((agg+x)@Wa + ba) @ Wb + bb )
// (includes the inter-layer ReLU). One 16-node tile per wave.
// ---------------------------------------------------------------------------
__global__ __launch_bounds__(256) void gin_mlp_mid(
    const float* __restrict__ x, const float* __restrict__ agg,
    const float* __restrict__ Wa, const float* __restrict__ ba,
    const float* __restrict__ Wb, const float* __restrict__ bb,
    float* __restrict__ h) {
  __shared__ float sWa[D * D];
  __shared__ float sWb[D * D];
  __shared__ float stage[WAVES * 16 * D];

  int tid = threadIdx.x;
  for (int i = tid; i < D * D; i += 256) { sWa[i] = Wa[i]; sWb[i] = Wb[i]; }
  __syncthreads();

  int wid = tid >> 5, lane = tid & 31;
  int tile = blockIdx.x * WAVES + wid;
  if (tile >= N_TILES) return;

  int m = lane & 15;          // row within tile (A) / column (B,C,D)
  int half = lane >> 4;       // K sub-offset select
  long rowbase = (long)(tile * 16 + m) * D;
  float* st = stage + wid * 16 * D;

  // ---- GEMM1: z @ Wa, z = agg + x, accumulated over K=64 in steps of 4 ----
  v8f acc[4] = {v8f{0}, v8f{0}, v8f{0}, v8f{0}};
#pragma unroll
  for (int s = 0; s < 16; ++s) {
    int k0 = 4 * s + 2 * half;
    v2f a  = *(const v2f*)(agg + rowbase + k0);
    v2f ax = *(const v2f*)(x + rowbase + k0);
    a.x += ax.x; a.y += ax.y;
#pragma unroll
    for (int t = 0; t < 4; ++t) {
      v2f b;
      b.x = sWa[k0 * D + 16 * t + m];
      b.y = sWa[(k0 + 1) * D + 16 * t + m];
      acc[t] = wmma_f32(a, b, acc[t]);
    }
  }

  // bias + ReLU, restage 16x64 activation tile into LDS (row-major)
#pragma unroll
  for (int t = 0; t < 4; ++t) {
    float bv = ba[16 * t + m];
#pragma unroll
    for (int r = 0; r < 8; ++r) {
      float v = acc[t][r] + bv;
      v = v > 0.f ? v : 0.f;
      st[(r + 8 * half) * D + 16 * t + m] = v;
    }
  }
  asm volatile("s_wait_dscnt 0" ::: "memory");  // per-wave LDS RAW fence

  // ---- GEMM2: h1 @ Wb ----
  v8f acc2[4] = {v8f{0}, v8f{0}, v8f{0}, v8f{0}};
#pragma unroll
  for (int s = 0; s < 16; ++s) {
    int k0 = 4 * s + 2 * half;
    v2f a = *(const v2f*)(st + m * D + k0);
#pragma unroll
    for (int t = 0; t < 4; ++t) {
      v2f b;
      b.x = sWb[k0 * D + 16 * t + m];
      b.y = sWb[(k0 + 1) * D + 16 * t + m];
      acc2[t] = wmma_f32(a, b, acc2[t]);
    }
  }

  // bias + inter-layer ReLU, store h
#pragma unroll
  for (int t = 0; t < 4; ++t) {
    float bv = bb[16 * t + m];
#pragma unroll
    for (int r = 0; r < 8; ++r) {
      float v = acc2[t][r] + bv;
      v = v > 0.f ? v : 0.f;
      h[(long)(tile * 16 + r + 8 * half) * D + 16 * t + m] = v;
    }
  }
}

// ---------------------------------------------------------------------------
// Layer 2 fused MLP + log_softmax:
// out = log_softmax( relu((agg+h)@Wa + ba) @ Wb + bb )
// Wb is 64x40, padded to 64x48 in LDS so N-tiling is 3 x 16.
// ---------------------------------------------------------------------------
__global__ __launch_bounds__(256) void gin_mlp_out(
    const float* __restrict__ hin, const float* __restrict__ agg,
    const float* __restrict__ Wa, const float* __restrict__ ba,
    const float* __restrict__ Wb, const float* __restrict__ bb,
    float* __restrict__ out) {
  __shared__ float sWa[D * D];
  __shared__ float sWb[D * D_OUT_PAD];
  __shared__ float stage[WAVES * 16 * D];

  int tid = threadIdx.x;
  for (int i = tid; i < D * D; i += 256) sWa[i] = Wa[i];
  for (int i = tid; i < D * D_OUT_PAD; i += 256) {
    int k = i / D_OUT_PAD, n = i % D_OUT_PAD;
    sWb[i] = (n < D_OUT) ? Wb[k * D_OUT + n] : 0.f;
  }
  __syncthreads();

  int wid = tid >> 5, lane = tid & 31;
  int tile = blockIdx.x * WAVES + wid;
  if (tile >= N_TILES) return;

  int m = lane & 15;
  int half = lane >> 4;
  long rowbase = (long)(tile * 16 + m) * D;
  float* st = stage + wid * 16 * D;

  // ---- GEMM1: (agg + h) @ Wa ----
  v8f acc[4] = {v8f{0}, v8f{0}, v8f{0}, v8f{0}};
#pragma unroll
  for (int s = 0; s < 16; ++s) {
    int k0 = 4 * s + 2 * half;
    v2f a  = *(const v2f*)(agg + rowbase + k0);
    v2f ah = *(const v2f*)(hin + rowbase + k0);
    a.x += ah.x; a.y += ah.y;
#pragma unroll
    for (int t = 0; t < 4; ++t) {
      v2f b;
      b.x = sWa[k0 * D + 16 * t + m];
      b.y = sWa[(k0 + 1) * D + 16 * t + m];
      acc[t] = wmma_f32(a, b, acc[t]);
    }
  }
#pragma unroll
  for (int t = 0; t < 4; ++t) {
    float bv = ba[16 * t + m];
#pragma unroll
    for (int r = 0; r < 8; ++r) {
      float v = acc[t][r] + bv;
      v = v > 0.f ? v : 0.f;
      st[(r + 8 * half) * D + 16 * t + m] = v;
    }
  }
  asm volatile("s_wait_dscnt 0" ::: "memory");

  // ---- GEMM2: h1 @ Wb (padded to 48 cols) ----
  v8f acc2[3] = {v8f{0}, v8f{0}, v8f{0}};
#pragma unroll
  for (int s = 0; s < 16; ++s) {
    int k0 = 4 * s + 2 * half;
    v2f a = *(const v2f*)(st + m * D + k0);
#pragma unroll
    for (int t = 0; t < 3; ++t) {
      v2f b;
      b.x = sWb[k0 * D_OUT_PAD + 16 * t + m];
      b.y = sWb[(k0 + 1) * D_OUT_PAD + 16 * t + m];
      acc2[t] = wmma_f32(a, b, acc2[t]);
    }
  }
  asm volatile("s_wait_dscnt 0" ::: "memory");  // ensure stage reads done before reuse

  // bias, write logits tile [16][48] into LDS
#pragma unroll
  for (int t = 0; t < 3; ++t) {
    int col = 16 * t + m;
    float bv = (col < D_OUT) ? bb[col] : 0.f;
#pragma unroll
    for (int r = 0; r < 8; ++r)
      st[(r + 8 * half) * D_OUT_PAD + col] = acc2[t][r] + bv;
  }
  asm volatile("s_wait_dscnt 0" ::: "memory");

  // log_softmax: lanes 0..15 each reduce one row of 40 logits
  if (half == 0) {
    const float* row = st + m * D_OUT_PAD;
    float mx = row[0];
#pragma unroll
    for (int j = 1; j < D_OUT; ++j) mx = row[j] > mx ? row[j] : mx;
    float ssum = 0.f;
#pragma unroll
    for (int j = 0; j < D_OUT; ++j) ssum += __expf(row[j] - mx);
    float lse = __logf(ssum);
    long ob = (long)(tile * 16 + m) * D_OUT;
#pragma unroll
    for (int j = 0; j < D_OUT; ++j) out[ob + j] = row[j] - mx - lse;
  }
}

// ---------------------------------------------------------------------------
extern "C" void kernel_launch(void* const* d_in, const int* in_sizes, int n_in,
                              void* d_out, int out_size, void* d_ws, size_t ws_size,
                              hipStream_t stream) {
  const float* x   = (const float*)d_in[0];
  const int*   ei  = (const int*)d_in[1];
  const float* W1a = (const float*)d_in[2];
  const float* b1a = (const float*)d_in[3];
  const float* W1b = (const float*)d_in[4];
  const float* b1b = (const float*)d_in[5];
  const float* W2a = (const float*)d_in[6];
  const float* b2a = (const float*)d_in[7];
  const float* W2b = (const float*)d_in[8];
  const float* b2b = (const float*)d_in[9];
  float* outp = (float*)d_out;

  const int* src = ei;
  const int* dst = ei + N_EDGES;

  float* agg = (float*)d_ws;                       // 100000*64 f32 = 25.6 MB
  float* h   = agg + (size_t)N_NODES * D;          // 25.6 MB

  const size_t featBytes = (size_t)N_NODES * D * sizeof(float);
  const int scatterBlocks = (N_EDGES * 32) / 256;  // 156250
  const int mlpBlocks = (N_TILES + WAVES - 1) / WAVES;  // 782

  // Layer 1
  hipMemsetAsync(agg, 0, featBytes, stream);
  gin_scatter<<<scatterBlocks, 256, 0, stream>>>(x, src, dst, agg);
  gin_mlp_mid<<<mlpBlocks, 256, 0, stream>>>(x, agg, W1a, b1a, W1b, b1b, h);

  // Layer 2
  hipMemsetAsync(agg, 0, featBytes, stream);
  gin_scatter<<<scatterBlocks, 256, 0, stream>>>(h, src, dst, agg);
  gin_mlp_out<<<mlpBlocks, 256, 0, stream>>>(h, agg, W2a, b2a, W2b, b2b, outp);
}